// DSSCSAB_49177375539284
// MI455X (gfx1250) — compile-verified
//
#include <hip/hip_runtime.h>
#include <hip/hip_bf16.h>
#include <math.h>

// ---------------------------------------------------------------- constants
#define NCH   56
#define NCONV 28            // SCA conv channels / FA v channels
#define NB_   8
#define HW_   192
#define LOCS  294912        // 8*192*192
#define SLICE 36864         // 192*192
#define NWIN  576           // 24*24 windows of 8x8
#define NTOK  64
#define DHEAD 14
#define FFNC  224

typedef __attribute__((ext_vector_type(2))) float v2f;
typedef __attribute__((ext_vector_type(8))) float v8f;

#ifndef __has_builtin
#define __has_builtin(x) 0
#endif
#if __has_builtin(__builtin_amdgcn_wmma_f32_16x16x4_f32)
#define HAVE_WMMA_F32X4 1
#else
#define HAVE_WMMA_F32X4 0
#endif

// Generic fp32 WMMA GEMM over 16x16 C tiles, K in steps of 4.
// A: MxK row-major (LDS, zero-padded, rows 8B-aligned), B: KxN (or NxK if
// transB), C row-major. Tiles distributed across the block's 4 waves; all
// lanes active (EXEC full) -- OOB columns are handled by clamping the index
// (always in-bounds) and zero-selecting the value, never by branching.
__device__ __forceinline__ void gemm_f32(const float* __restrict__ A, int lda,
                                         const float* __restrict__ B, int ldb,
                                         bool transB, float* __restrict__ C, int ldc,
                                         int Mt, int Nt, int Kt, int nvalid,
                                         int wave, int lane)
{
  __builtin_assume(A != nullptr);
  __builtin_assume(B != nullptr);
  __builtin_assume(C != nullptr);
  for (int t = wave; t < Mt * Nt; t += 4) {
    const int m0 = (t / Nt) * 16;
    const int n0 = (t % Nt) * 16;
    v8f acc = {0.f, 0.f, 0.f, 0.f, 0.f, 0.f, 0.f, 0.f};
#if HAVE_WMMA_F32X4
    const int m = m0 + (lane & 15);
    const int n = n0 + (lane & 15);
    const int khalf = (lane >> 4) << 1;
    const bool nok = (n < nvalid);
    const int nc = nok ? n : 0;          // clamped, always in-bounds
    for (int ks = 0; ks < Kt; ++ks) {
      const int kk = ks * 4 + khalf;     // even -> 8B aligned pair
      const v2f a = *(const v2f*)(A + m * lda + kk);
      v2f b;
      if (transB) {
        const v2f bt = *(const v2f*)(B + nc * ldb + kk);
        b.x = nok ? bt.x : 0.f;
        b.y = nok ? bt.y : 0.f;
      } else {
        b.x = nok ? B[kk * ldb + nc]       : 0.f;
        b.y = nok ? B[(kk + 1) * ldb + nc] : 0.f;
      }
      acc = __builtin_amdgcn_wmma_f32_16x16x4_f32(false, a, false, b,
                                                  (short)0, acc, false, false);
    }
    for (int r = 0; r < 8; ++r)
      C[(m0 + r + ((lane >> 4) << 3)) * ldc + n0 + (lane & 15)] = acc[r];
#else
    for (int r = 0; r < 8; ++r) {
      const int row = m0 + r + ((lane >> 4) << 3);
      const int col = n0 + (lane & 15);
      const int cc = (col < nvalid) ? col : 0;
      float s = 0.f;
      for (int k = 0; k < Kt * 4; ++k) {
        float bv = (col < nvalid) ? (transB ? B[cc * ldb + k] : B[k * ldb + cc]) : 0.f;
        s += A[row * lda + k] * bv;
      }
      C[row * ldc + col] = s;
    }
#endif
  }
}

__device__ __forceinline__ int toff(int n) { return (n >> 3) * HW_ + (n & 7); }

// Attention core (2 heads, d=14, 64 tokens). qp/kp: [64][32] (cols 16h..16h+13
// valid, rest zero). vv: [64][32] (cols 0..27 valid, rest zero).
// sim: [64][64] scratch. out: [64][32] (cols 0..27 produced). vec: >=257 floats.
__device__ void attn_core(float* qp, float* kp, float* vv, float* sim, float* out,
                          float* vec,
                          const float* pcqw, const float* pcqb,
                          const float* pckw, const float* pckb,
                          const float* m1, const float* m2a, const float* m2b,
                          int tid)
{
  __builtin_assume(qp != nullptr);
  __builtin_assume(kp != nullptr);
  __builtin_assume(vv != nullptr);
  __builtin_assume(sim != nullptr);
  __builtin_assume(out != nullptr);
  __builtin_assume(vec != nullptr);
  const int lane = tid & 31, wave = tid >> 5;
  float* Sq = vec;        float* Sk = vec + 64;
  float* t1 = vec + 128;  float* t2 = vec + 192;
  float* th = vec + 256;
  for (int h = 0; h < 2; ++h) {
    // sim = q_h (64x14) @ k_h^T
    gemm_f32(qp + 16 * h, 32, kp + 16 * h, 32, true, sim, 64, 4, 4, 4, 1024, wave, lane);
    __syncthreads();
    if (tid < 64) {
      float sq = 0.f, sk = 0.f;
      for (int d = 0; d < DHEAD; ++d) {
        sq += qp[tid * 32 + 16 * h + d] * pcqw[d];
        sk += kp[tid * 32 + 16 * h + d] * pckw[d];
      }
      Sq[tid] = sq + pcqb[0];
      Sk[tid] = sk + pckb[0];
      float acc = 0.f;                        // t = (sim, diag removed) . m1
      for (int j = 0; j < NTOK; ++j) acc += sim[tid * 64 + j] * m1[j];
      acc -= sim[tid * 64 + tid] * m1[tid];
      t1[tid] = acc;
    }
    __syncthreads();
    if (tid < 64) {                           // leaky_relu(m2a @ t, 0.1)
      float acc = 0.f;
      for (int i = 0; i < NTOK; ++i) acc += m2a[tid * 64 + i] * t1[i];
      t2[tid] = acc > 0.f ? acc : 0.1f * acc;
    }
    __syncthreads();
    if (tid == 0) {
      float acc = 0.f;
      for (int j = 0; j < NTOK; ++j) acc += t2[j] * m2b[j];
      th[0] = acc;
    }
    __syncthreads();
    if (tid < 64) {                           // sims = sim*Sigma; softmax * (sims>theta)
      const float theta = th[0], sqi = Sq[tid];
      float mx = -3.4e38f;
      for (int j = 0; j < NTOK; ++j) {
        float s = sim[tid * 64 + j] * sqi * Sk[j];
        sim[tid * 64 + j] = s;
        mx = s > mx ? s : mx;
      }
      float sum = 0.f;
      for (int j = 0; j < NTOK; ++j) {
        float s = sim[tid * 64 + j];
        float e = expf(s - mx);
        sum += e;                             // denominator over ALL j
        sim[tid * 64 + j] = (s > theta) ? e : 0.f;  // masked numerator
      }
      float inv = 1.f / sum;
      for (int j = 0; j < NTOK; ++j) sim[tid * 64 + j] *= inv;
    }
    __syncthreads();
    // out_h = attn (64x64) @ v_h (64x14); head1 overwrites head0's pad cols 14,15
    gemm_f32(sim, 64, vv + DHEAD * h, 32, false, out + DHEAD * h, 32, 4, 1, 16, 1024,
             wave, lane);
    __syncthreads();
  }
}

// ---------------------------------------------------------------- FA block
__global__ __launch_bounds__(128)
void fa_kernel(const float* __restrict__ x,
               const float* __restrict__ lnw, const float* __restrict__ lnb,
               const float* __restrict__ qkw, const float* __restrict__ vw,
               const float* __restrict__ ow,  const float* __restrict__ ob,
               const float* pcqw, const float* pcqb,
               const float* pckw, const float* pckb,
               const float* m1, const float* m2a, const float* m2b,
               float* __restrict__ x1)
{
  __shared__ float s_a[64 * 64];   // xw -> later qpad(2048)+kpad(2048)
  __shared__ float s_b[64 * 64];   // qk proj -> later out[64][32]
  __shared__ float s_v[64 * 32];
  __shared__ float s_sim[64 * 64]; // sim / final projection result
  __shared__ float s_vec[5 * 64];
  const int tid = threadIdx.x, lane = tid & 31, wave = tid >> 5;
  const int wi = blockIdx.x, Bi = blockIdx.y;
  const int base = Bi * SLICE + (wi / 24) * 8 * HW_ + (wi % 24) * 8;

  if (tid < 64) {                       // LayerNorm over channels per token
    const int off = base + toff(tid);
    float sum = 0.f, sq = 0.f;
    for (int c = 0; c < NCH; ++c) {
      float v = x[c * LOCS + off];
      s_a[tid * 64 + c] = v;
      sum += v; sq += v * v;
    }
    const float mu = sum * (1.f / NCH);
    const float inv = rsqrtf(sq * (1.f / NCH) - mu * mu + 1e-5f);
    for (int c = 0; c < NCH; ++c)
      s_a[tid * 64 + c] = (s_a[tid * 64 + c] - mu) * inv * lnw[c] + lnb[c];
  }
  __syncthreads();
  // v = xw @ vw^T  (64x28, K=56);  qk = xw @ qkw^T (64x56)
  gemm_f32(s_a, 64, vw, NCH, true, s_v, 32, 4, 2, 14, NCONV, wave, lane);
  gemm_f32(s_a, 64, qkw, NCH, true, s_b, 64, 4, 4, 14, NCH, wave, lane);
  __syncthreads();
  // build zero-padded per-head q,k in s_a  (q: s_b cols 0..27, k: 28..55)
  for (int i = tid; i < 64 * 32; i += 128) {
    const int n = i >> 5, col = i & 31, h = col >> 4, j = col & 15;
    s_a[i]        = (j < DHEAD) ? s_b[n * 64 + h * DHEAD + j] : 0.f;
    s_a[2048 + i] = (j < DHEAD) ? s_b[n * 64 + NCONV + h * DHEAD + j] : 0.f;
  }
  __syncthreads();
  attn_core(s_a, s_a + 2048, s_v, s_sim, s_b, s_vec,
            pcqw, pcqb, pckw, pckb, m1, m2a, m2b, tid);
  // final projection: out (64x28) @ ow^T (56x28) -> s_sim[64][64]
  gemm_f32(s_b, 32, ow, NCONV, true, s_sim, 64, 4, 4, 7, NCH, wave, lane);
  __syncthreads();
  for (int e = tid; e < NTOK * NCH; e += 128) {
    const int n = e / NCH, o = e % NCH;
    const int gi = o * LOCS + base + toff(n);
    x1[gi] = s_sim[n * 64 + o] + ob[o] + x[gi];
  }
}

// ---------------------------------------------------------------- channel LN
__global__ void ln_kernel(const float* __restrict__ src,
                          const float* __restrict__ w, const float* __restrict__ b,
                          float* __restrict__ dst)
{
  const int loc = blockIdx.x * blockDim.x + threadIdx.x;
  float sum = 0.f, sq = 0.f;
  for (int c = 0; c < NCH; ++c) {
    float v = src[c * LOCS + loc];
    sum += v; sq += v * v;
  }
  const float mu = sum * (1.f / NCH);
  const float inv = rsqrtf(sq * (1.f / NCH) - mu * mu + 1e-5f);
  for (int c = 0; c < NCH; ++c)
    dst[c * LOCS + loc] = (src[c * LOCS + loc] - mu) * inv * w[c] + b[c];
}

// ---------------------------------------------------------------- SCA convs
__global__ void sca_conv_kernel(const float* __restrict__ lnA,
                                const float* __restrict__ lnB,
                                const float* __restrict__ qw,
                                const float* __restrict__ kw,
                                const float* __restrict__ vw,
                                float* __restrict__ qc, float* __restrict__ kc,
                                float* __restrict__ vc)
{
  const int which = blockIdx.y;
  const int idx = blockIdx.x * 256 + threadIdx.x;      // < 28*LOCS
  const int g = idx / LOCS, loc = idx % LOCS;
  const int Bi = loc / SLICE, r = loc % SLICE, h = r / HW_, w = r % HW_;
  const float* src = (which == 0) ? lnA : lnB;
  const float* W = (which == 0) ? qw : (which == 1 ? kw : vw);
  float acc = 0.f;
  for (int ic = 0; ic < 2; ++ic) {
    const int cin = g * 2 + ic;
    const float* wp = W + (g * 2 + ic) * 27;
    for (int db = -1; db <= 1; ++db) {
      const int Bn = Bi + db;
      for (int dy = -1; dy <= 1; ++dy) {
        const int hn = h + dy;
        for (int dx = -1; dx <= 1; ++dx) {
          const int wn = w + dx;
          float v = (Bn >= 0 && Bn < NB_ && hn >= 0 && hn < HW_ && wn >= 0 && wn < HW_)
                        ? src[cin * LOCS + Bn * SLICE + hn * HW_ + wn] : 0.f;
          acc += v * wp[(db + 1) * 9 + (dy + 1) * 3 + (dx + 1)];
        }
      }
    }
  }
  float* dst = (which == 0) ? qc : (which == 1 ? kc : vc);
  dst[idx] = acc;
}

// ---------------------------------------------------------------- SCA block
__global__ __launch_bounds__(128)
void sca_kernel(const float* __restrict__ qc, const float* __restrict__ kc,
                const float* __restrict__ vc, const float* __restrict__ x1,
                const float* __restrict__ sow,
                const float* pcqw, const float* pcqb,
                const float* pckw, const float* pckb,
                const float* m1, const float* m2a, const float* m2b,
                float* __restrict__ x2)
{
  __shared__ float s_qp[64 * 32];
  __shared__ float s_kp[64 * 32];
  __shared__ float s_v[64 * 32];
  __shared__ float s_sim[64 * 64];
  __shared__ float s_out[64 * 32];
  __shared__ float s_vec[5 * 64];
  const int tid = threadIdx.x;
  const int wi = blockIdx.x, Bi = blockIdx.y;
  const int base = Bi * SLICE + (wi / 24) * 8 * HW_ + (wi % 24) * 8;

  for (int i = tid; i < 64 * 32; i += 128) {
    const int n = i >> 5, col = i & 31, h = col >> 4, j = col & 15;
    const int off = base + toff(n);
    s_qp[i] = (j < DHEAD) ? qc[(h * DHEAD + j) * LOCS + off] : 0.f;
    s_kp[i] = (j < DHEAD) ? kc[(h * DHEAD + j) * LOCS + off] : 0.f;
    s_v[i]  = (col < NCONV) ? vc[col * LOCS + off] : 0.f;
  }
  __syncthreads();
  attn_core(s_qp, s_kp, s_v, s_sim, s_out, s_vec,
            pcqw, pcqb, pckw, pckb, m1, m2a, m2b, tid);
  // 1x1x1 grouped out-conv (2 outputs per input channel) + residual
  for (int e = tid; e < NTOK * NCONV; e += 128) {
    const int n = e / NCONV, c = e % NCONV;
    const float val = s_out[n * 32 + c];
    const int off = base + toff(n);
    const int i0 = (2 * c) * LOCS + off, i1 = (2 * c + 1) * LOCS + off;
    x2[i0] = val * sow[2 * c] + x1[i0];
    x2[i1] = val * sow[2 * c + 1] + x1[i1];
  }
}

// ---------------------------------------------------------------- FFN conv1 (WMMA)
__global__ __launch_bounds__(128)
void ffn1_kernel(const float* __restrict__ lnA, const float* __restrict__ lnB,
                 const float* __restrict__ w1x, const float* __restrict__ w1l,
                 float* __restrict__ hA, float* __restrict__ hB, int Bi)
{
  __shared__ float s_a[16 * 64];
  const int path = blockIdx.y;
  const float* src = path ? lnB : lnA;
  const float* W = path ? w1l : w1x;
  float* H = path ? hB : hA;
  const int tid = threadIdx.x, lane = tid & 31, wave = tid >> 5;
  const int base = Bi * SLICE + blockIdx.x * 16;
  for (int i = tid; i < 16 * 64; i += 128) {
    const int m = i >> 6, c = i & 63;
    s_a[i] = (c < NCH) ? src[c * LOCS + base + m] : 0.f;
  }
  __syncthreads();
  // h[m][oc] = sum_c A[m][c] * W[oc][c] ; M=16, N=224, K=56
  gemm_f32(s_a, 64, W, NCH, true, H + (size_t)blockIdx.x * 16 * FFNC, FFNC,
           1, 14, 14, FFNC, wave, lane);
}

// ---------------------------------------------------------------- FFN conv2+gate
__global__ void ffn2_kernel(const float* __restrict__ hA, const float* __restrict__ hB,
                            const float* __restrict__ w2x, const float* __restrict__ w2l,
                            const float* __restrict__ x2, float* __restrict__ out,
                            int Bi)
{
  const int hw = blockIdx.x * 256 + threadIdx.x;   // < SLICE
  const int o = blockIdx.y;
  const int h = hw / HW_, w = hw % HW_;
  float xb = 0.f, lb = 0.f;
  for (int i = 0; i < 4; ++i) {
    for (int dy = -1; dy <= 1; ++dy) {
      const int hn = h + dy;
      if (hn < 0 || hn >= HW_) continue;
      for (int dx = -1; dx <= 1; ++dx) {
        const int wn = w + dx;
        if (wn < 0 || wn >= HW_) continue;
        const int widx = (o * 4 + i) * 9 + (dy + 1) * 3 + (dx + 1);
        const int hi = (hn * HW_ + wn) * FFNC + o * 4 + i;
        xb += hA[hi] * w2x[widx];
        lb += hB[hi] * w2l[widx];
      }
    }
  }
  const float g = 0.5f * lb * (1.f + erff(lb * 0.70710678118654752f));  // exact GELU
  const int gi = o * LOCS + Bi * SLICE + hw;
  out[gi] = xb * g + x2[gi];
}

// ---------------------------------------------------------------- launcher
extern "C" void kernel_launch(void* const* d_in, const int* in_sizes, int n_in,
                              void* d_out, int out_size, void* d_ws, size_t ws_size,
                              hipStream_t stream)
{
  const float* x        = (const float*)d_in[0];
  const float* last     = (const float*)d_in[1];
  const float* fa_lnw   = (const float*)d_in[2];
  const float* fa_lnb   = (const float*)d_in[3];
  const float* fa_qkw   = (const float*)d_in[4];
  const float* fa_vw    = (const float*)d_in[5];
  const float* fa_ow    = (const float*)d_in[6];
  const float* fa_ob    = (const float*)d_in[7];
  const float* fa_pcqw  = (const float*)d_in[8];
  const float* fa_pcqb  = (const float*)d_in[9];
  const float* fa_pckw  = (const float*)d_in[10];
  const float* fa_pckb  = (const float*)d_in[11];
  const float* fa_m1    = (const float*)d_in[12];
  const float* fa_m2a   = (const float*)d_in[13];
  const float* fa_m2b   = (const float*)d_in[14];
  const float* s_ln1w   = (const float*)d_in[15];
  const float* s_ln1b   = (const float*)d_in[16];
  const float* s_ln2w   = (const float*)d_in[17];
  const float* s_ln2b   = (const float*)d_in[18];
  const float* s_qw     = (const float*)d_in[19];
  const float* s_kw     = (const float*)d_in[20];
  const float* s_vw     = (const float*)d_in[21];
  const float* s_ow     = (const float*)d_in[22];
  const float* s_pcqw   = (const float*)d_in[23];
  const float* s_pcqb   = (const float*)d_in[24];
  const float* s_pckw   = (const float*)d_in[25];
  const float* s_pckb   = (const float*)d_in[26];
  const float* s_m1     = (const float*)d_in[27];
  const float* s_m2a    = (const float*)d_in[28];
  const float* s_m2b    = (const float*)d_in[29];
  const float* f_ln1w   = (const float*)d_in[30];
  const float* f_ln1b   = (const float*)d_in[31];
  const float* f_ln2w   = (const float*)d_in[32];
  const float* f_ln2b   = (const float*)d_in[33];
  const float* f_x1w    = (const float*)d_in[34];
  const float* f_x2w    = (const float*)d_in[35];
  const float* f_l1w    = (const float*)d_in[36];
  const float* f_l2w    = (const float*)d_in[37];
  float* out = (float*)d_out;
  float* ws = (float*)d_ws;

  const size_t T  = (size_t)NCH * LOCS;     // 16,515,072
  const size_t Tc = (size_t)NCONV * LOCS;   //  8,257,536
  float* x1  = ws;            // post-FA x ; later reused as FFN hidden (hA|hB)
  float* x2  = ws + T;        // post-SCA x
  float* lnA = ws + 2 * T;
  float* lnB = ws + 3 * T;
  float* qc  = ws + 4 * T;
  float* kc  = qc + Tc;
  float* vc  = kc + Tc;
  float* hA  = x1;            // 224*SLICE per B-slice == Tc
  float* hB  = x1 + Tc;

  // --- FA ---
  fa_kernel<<<dim3(NWIN, NB_), 128, 0, stream>>>(
      x, fa_lnw, fa_lnb, fa_qkw, fa_vw, fa_ow, fa_ob,
      fa_pcqw, fa_pcqb, fa_pckw, fa_pckb, fa_m1, fa_m2a, fa_m2b, x1);
  // --- SCA ---
  ln_kernel<<<LOCS / 256, 256, 0, stream>>>(x1, s_ln1w, s_ln1b, lnA);
  ln_kernel<<<LOCS / 256, 256, 0, stream>>>(last, s_ln2w, s_ln2b, lnB);
  sca_conv_kernel<<<dim3((NCONV * LOCS) / 256, 3), 256, 0, stream>>>(
      lnA, lnB, s_qw, s_kw, s_vw, qc, kc, vc);
  sca_kernel<<<dim3(NWIN, NB_), 128, 0, stream>>>(
      qc, kc, vc, x1, s_ow,
      s_pcqw, s_pcqb, s_pckw, s_pckb, s_m1, s_m2a, s_m2b, x2);
  // --- FFN ---
  ln_kernel<<<LOCS / 256, 256, 0, stream>>>(x2, f_ln1w, f_ln1b, lnA);
  ln_kernel<<<LOCS / 256, 256, 0, stream>>>(last, f_ln2w, f_ln2b, lnB);
  for (int Bi = 0; Bi < NB_; ++Bi) {
    ffn1_kernel<<<dim3(SLICE / 16, 2), 128, 0, stream>>>(
        lnA, lnB, f_x1w, f_l1w, hA, hB, Bi);
    ffn2_kernel<<<dim3(SLICE / 256, NCH), 256, 0, stream>>>(
        hA, hB, f_x2w, f_l2w, x2, out, Bi);
  }
}